// vMF_455266533961
// MI455X (gfx1250) — compile-verified
//
#include <hip/hip_runtime.h>

// ---------------------------------------------------------------------------
// vMF log-likelihood for MI455X (gfx1250, wave32, WMMA).
//   out[i] = kappa * dot(x[i], mu_hat) + logC
// Dominated by streaming x (1 GiB @ 23.3 TB/s -> ~46us floor). fp32 kept for
// accuracy; V_WMMA_F32_16X16X4_F32 performs the 16-row dot-product reduction
// with a broadcast-B trick at zero extra bandwidth.
// ---------------------------------------------------------------------------

typedef __attribute__((ext_vector_type(2))) float v2f;
typedef __attribute__((ext_vector_type(8))) float v8f;

#define D_DIM            512
#define ROWS_PER_WAVE    16
#define WAVES_PER_BLOCK  8
#define ROWS_PER_BLOCK   (ROWS_PER_WAVE * WAVES_PER_BLOCK)   // 128
#define CHUNK_COLS       64
#define NUM_CHUNKS       (D_DIM / CHUNK_COLS)                // 8
#define TILE_STRIDE      68   // 64 cols + 4 pad floats: row m starts at bank 4m
                              // -> conflict-free WMMA-A-layout float2 reads

// ---------------------------------------------------------------------------
// Scalar prep: norm(mu), kappa, logC via 512-term log I_s(kappa) series.
// One block of 512 threads; double precision (cost is negligible).
// ws[0..511] = kappa * mu / ||mu||,  ws[512] = logC
// ---------------------------------------------------------------------------
__global__ void vmf_prep_kernel(const float* __restrict__ mu_unnorm,
                                const float* __restrict__ logkappa,
                                float* __restrict__ ws) {
  __shared__ double red[D_DIM];
  const int t = threadIdx.x;
  const float m = mu_unnorm[t];

  // ||mu||^2
  red[t] = (double)m * (double)m;
  __syncthreads();
  for (int o = 256; o > 0; o >>= 1) {
    if (t < o) red[t] += red[t + o];
    __syncthreads();
  }
  const double norm = sqrt(red[0]);
  __syncthreads();

  const double kappa = exp((double)logkappa[0]) + 1e-6;
  const double s = 0.5 * (double)D_DIM - 1.0;   // 255
  const double k = (double)t;
  // log term_k = (2k+s)*log(kappa/2) - lgamma(k+1) - lgamma(k+s+1)
  const double lt = (2.0 * k + s) * log(kappa * 0.5)
                    - lgamma(k + 1.0) - lgamma(k + s + 1.0);

  // logsumexp over the 512 terms
  red[t] = lt;
  __syncthreads();
  for (int o = 256; o > 0; o >>= 1) {
    if (t < o) red[t] = fmax(red[t], red[t + o]);
    __syncthreads();
  }
  const double mx = red[0];
  __syncthreads();
  red[t] = exp(lt - mx);
  __syncthreads();
  for (int o = 256; o > 0; o >>= 1) {
    if (t < o) red[t] += red[t + o];
    __syncthreads();
  }
  const double logI = mx + log(red[0]);

  // logC = d * (-0.5*log(2*pi)) + s*log(kappa) - logI
  const double logC =
      (double)D_DIM * (-0.91893853320467274178) + s * log(kappa) - logI;

  ws[t] = (float)(kappa * (double)m / norm);
  if (t == 0) ws[D_DIM] = (float)logC;
}

// ---------------------------------------------------------------------------
// GEMV: each wave owns 16 rows. Coalesced b128 global loads -> padded LDS
// tile -> WMMA-A-layout float2 reads. B[k][n] = w[k] broadcast across all 16
// columns, so C[m][n] = dot(row m, w) for every n. 128 WMMA per row-tile.
// ---------------------------------------------------------------------------
__global__ void __launch_bounds__(256)
vmf_gemv_wmma_kernel(const float* __restrict__ x,
                     const float* __restrict__ ws,
                     float* __restrict__ out,
                     int nrows) {
  __shared__ float wlds[D_DIM];
  __shared__ float tile[WAVES_PER_BLOCK][ROWS_PER_WAVE * TILE_STRIDE];

  const int tid  = threadIdx.x;
  const int lane = tid & 31;
  const int wv   = tid >> 5;
  const int m    = lane & 15;   // row within tile (A) / column (B,C)
  const int h    = lane >> 4;   // half-wave select: K pair 0-1 vs 2-3

  for (int i = tid; i < D_DIM; i += 256) wlds[i] = ws[i];
  const float logC = ws[D_DIM];
  __syncthreads();

  const int rowBase = blockIdx.x * ROWS_PER_BLOCK + wv * ROWS_PER_WAVE;
  float* mytile = tile[wv];

  v8f acc = {0.f, 0.f, 0.f, 0.f, 0.f, 0.f, 0.f, 0.f};

  for (int ch = 0; ch < NUM_CHUNKS; ++ch) {
    const int c0 = ch * CHUNK_COLS;

    // Stage 16 rows x 64 cols into LDS. Each b128 covers 2 full rows:
    // lanes 0-15 -> row 2i cols c0+4m.., lanes 16-31 -> row 2i+1. Fully
    // coalesced (2 x 256B per instruction).
    #pragma unroll
    for (int i = 0; i < 8; ++i) {
      int r = rowBase + 2 * i + h;
      r = (r < nrows) ? r : (nrows - 1);   // clamp: EXEC must stay all-1 for WMMA
      const float4 v = *(const float4*)(x + (size_t)r * D_DIM + (c0 + 4 * m));
      *(float4*)&mytile[(2 * i + h) * TILE_STRIDE + 4 * m] = v;
    }

    // 16 WMMA steps, K advances by 4 each step.
    //   A (16x4 f32): lane m V0/V1 = x[m][4s],x[m][4s+1]; lane m+16 = +2,+3
    //   B (4x16 f32): V0 lanes0-15 = w[4s] (K=0), lanes16-31 = w[4s+2] (K=2);
    //                 V1 = w[4s+1] / w[4s+3]  -> broadcast across columns.
    #pragma unroll
    for (int s = 0; s < 16; ++s) {
      const float2 bb = *(const float2*)&wlds[c0 + 4 * s + 2 * h];
      const float2 aa = *(const float2*)&mytile[m * TILE_STRIDE + 4 * s + 2 * h];
      v2f a2; a2.x = aa.x; a2.y = aa.y;
      v2f b2; b2.x = bb.x; b2.y = bb.y;
      acc = __builtin_amdgcn_wmma_f32_16x16x4_f32(
          /*neg_a=*/false, a2, /*neg_b=*/false, b2,
          /*c_mod=*/(short)0, acc, /*reuse_a=*/false, /*reuse_b=*/false);
    }
  }

  // C/D layout (32-bit 16x16): lane n (0-15) VGPR v = C[v][n];
  // lane n+16 VGPR v = C[v+8][n]. All columns identical (broadcast B), so
  // lane 0 holds dots for rows 0..7 and lane 16 for rows 8..15.
  if (m == 0) {
    const int r0 = rowBase + h * 8;
    #pragma unroll
    for (int j = 0; j < 8; ++j) {
      const int r = r0 + j;
      if (r < nrows) out[r] = acc[j] + logC;
    }
  }
}

// ---------------------------------------------------------------------------
extern "C" void kernel_launch(void* const* d_in, const int* in_sizes, int n_in,
                              void* d_out, int out_size, void* d_ws, size_t ws_size,
                              hipStream_t stream) {
  const float* x  = (const float*)d_in[0];   // [N, 512] fp32, unit rows
  const float* mu = (const float*)d_in[1];   // [512] fp32
  const float* lk = (const float*)d_in[2];   // [1] fp32
  float* out = (float*)d_out;                // [N] fp32
  float* ws  = (float*)d_ws;                 // w[512] + logC

  const int d = in_sizes[1];                 // 512
  const int n = in_sizes[0] / d;             // 524288

  vmf_prep_kernel<<<1, D_DIM, 0, stream>>>(mu, lk, ws);

  const int blocks = (n + ROWS_PER_BLOCK - 1) / ROWS_PER_BLOCK;
  vmf_gemv_wmma_kernel<<<blocks, 256, 0, stream>>>(x, ws, out, n);
}